// EdgeBipartiteDenoiserV4_19000935317552
// MI455X (gfx1250) — compile-verified
//
#include <hip/hip_runtime.h>
#include <hip/hip_bf16.h>
#include <math.h>

typedef __attribute__((ext_vector_type(2))) float v2f;
typedef __attribute__((ext_vector_type(8))) float v8f;

#define HID 128
#define NH  8
#define HD  16

// ---------------------------------------------------------------------------
// WMMA 16x16 tile MAC: acc += A[rowBase:rowBase+16, 0:K] @ B[0:K, colBase:colBase+16]
// fp32 WMMA (V_WMMA_F32_16X16X4_F32), K stepped by 4.
// Layouts per CDNA5 ISA 7.12.2:
//   A 16x4 : lane(0..15)=row M, VGPR0/1 = K{0,1} (lanes 0-15) / K{2,3} (lanes 16-31)
//   B 4x16 : lane(0..15)=col N, VGPR0/1 = K{0,1} (lanes 0-15) / K{2,3} (lanes 16-31)
//   C 16x16: VGPR r = rows (r, r+8) split across lane halves, col = lane&15
// ---------------------------------------------------------------------------
__device__ inline v8f wmma_tile_mac(const float* A, int lda,
                                    const float* B, int ldb,
                                    int rowBase, int colBase, int K,
                                    int maxRow, v8f acc)
{
    const int lane = threadIdx.x & 31;
    const int half = lane >> 4;
    const int l15  = lane & 15;
    int aRow = rowBase + l15;
    if (aRow > maxRow) aRow = maxRow;   // clamp loads (stores guarded by caller)
    const float* aP = A + (size_t)aRow * lda + 2 * half;
    const float* bP = B + (size_t)(2 * half) * ldb + colBase + l15;
    for (int k0 = 0; k0 < K; k0 += 4) {
        v2f a = *(const v2f*)(aP + k0);
        v2f b;
        b.x = bP[(size_t)k0 * ldb];
        b.y = bP[(size_t)k0 * ldb + ldb];
        acc = __builtin_amdgcn_wmma_f32_16x16x4_f32(false, a, false, b,
                                                    (short)0, acc, false, false);
    }
    return acc;
}

// ---------------------------------------------------------------------------
// Generic C[M x 128] = A[M x K] @ B[K x 128].  256 threads = 8 waves, each wave
// owns one 16-col slice.  Optional per-column sum / sum-of-squares (for BN).
// ---------------------------------------------------------------------------
__global__ void gemm128_wmma(const float* __restrict__ A,
                             const float* __restrict__ B,
                             float* __restrict__ C,
                             int M, int K,
                             float* colsum, float* colsq)
{
    const int colBase = (threadIdx.x >> 5) * 16;
    const int rowBase = blockIdx.x * 16;
    const int lane = threadIdx.x & 31;
    const int half = lane >> 4;
    const int l15  = lane & 15;

    v8f acc = {};
    acc = wmma_tile_mac(A, K, B, HID, rowBase, colBase, K, M - 1, acc);

    union { v8f v; float f[8]; } u; u.v = acc;
    float s8 = 0.f, q8 = 0.f;
#pragma unroll
    for (int r = 0; r < 8; ++r) {
        const int row = rowBase + r + 8 * half;
        const float val = u.f[r];
        if (row < M) C[(size_t)row * HID + colBase + l15] = val;
        s8 += val; q8 += val * val;
    }
    if (colsum != nullptr) {
        // lanes l and l+16 hold the same column (different rows): combine.
        s8 += __shfl_xor(s8, 16);
        q8 += __shfl_xor(q8, 16);
        if (lane < 16) {
            atomicAdd(&colsum[colBase + l15], s8);
            atomicAdd(&colsq[colBase + l15],  q8);
        }
    }
}

// ---------------------------------------------------------------------------
// Workspace init: agg = 0, segSum = 0, segMax = -inf, col stats = 0.
// ---------------------------------------------------------------------------
__global__ void init_ws_kernel(float* agg, float* segMax, float* segSum,
                               float* colsum, float* colsq, int NR)
{
    const int idx = blockIdx.x * blockDim.x + threadIdx.x;
    if (idx < NR * HID) agg[idx] = 0.0f;
    if (idx < NR * NH) {
        segSum[idx] = 0.0f;
        segMax[idx] = -INFINITY;
    }
    if (idx < HID) { colsum[idx] = 0.0f; colsq[idx] = 0.0f; }
}

__device__ inline void atomicMaxFloat(float* addr, float val)
{
    if (val >= 0.0f) atomicMax((int*)addr, __float_as_int(val));
    else             atomicMin((unsigned int*)addr, __float_as_uint(val));
}

// ---------------------------------------------------------------------------
// Pass 1: per-edge logits[e,h] = sum_d q[dst,h,d]*(k[src,h,d]+ke[e,h,d]) / 4
// One wave per edge: lane = h*4 + dq, each lane covers 4 d-values (float4).
// ke recomputed on the fly from the 4 edge features (Wke staged in LDS).
// Also atomic-max into segMax[dst,h].
// ---------------------------------------------------------------------------
__global__ void edge_logits_kernel(const float* __restrict__ Q,
                                   const float* __restrict__ Kp,
                                   const float* __restrict__ EF,
                                   const int* __restrict__ esrc,
                                   const int* __restrict__ edst,
                                   const float* __restrict__ Wke,
                                   float* __restrict__ logitsW,
                                   float* __restrict__ segMax, int E)
{
    __shared__ float WkeS[4 * HID];
    for (int i = threadIdx.x; i < 4 * HID; i += blockDim.x) WkeS[i] = Wke[i];
    __syncthreads();

    const int wave = threadIdx.x >> 5;
    const int lane = threadIdx.x & 31;
    const int e = blockIdx.x * 8 + wave;
    if (e >= E) return;

    const int h  = lane >> 2;
    const int dq = lane & 3;
    const int c  = h * HD + dq * 4;

    const int src = esrc[e];
    const int dst = edst[e];
    const float e0 = EF[e * 4 + 0], e1 = EF[e * 4 + 1];
    const float e2 = EF[e * 4 + 2], e3 = EF[e * 4 + 3];

    const float4 q4 = *(const float4*)(Q  + (size_t)dst * HID + c);
    const float4 k4 = *(const float4*)(Kp + (size_t)src * HID + c);

    float p = 0.0f;
#pragma unroll
    for (int j = 0; j < 4; ++j) {
        const float ke = e0 * WkeS[c + j] + e1 * WkeS[HID + c + j]
                       + e2 * WkeS[2 * HID + c + j] + e3 * WkeS[3 * HID + c + j];
        p += (&q4.x)[j] * ((&k4.x)[j] + ke);
    }
    p += __shfl_xor(p, 1);
    p += __shfl_xor(p, 2);
    const float logit = p * 0.25f;   // 1/sqrt(HD)

    if (dq == 0) {
        logitsW[(size_t)e * NH + h] = logit;
        atomicMaxFloat(&segMax[(size_t)dst * NH + h], logit);
    }
}

// ---------------------------------------------------------------------------
// Pass 2: ex = exp(logit - segMax[dst]); segSum[dst] += ex.  One thread per (e,h).
// ---------------------------------------------------------------------------
__global__ void edge_exp_kernel(const float* __restrict__ logitsW,
                                const int* __restrict__ edst,
                                const float* __restrict__ segMax,
                                float* __restrict__ exW,
                                float* __restrict__ segSum, int E)
{
    const int idx = blockIdx.x * blockDim.x + threadIdx.x;
    if (idx >= E * NH) return;
    const int e = idx >> 3;
    const int h = idx & 7;
    const int dst = edst[e];
    const float ex = expf(logitsW[idx] - segMax[(size_t)dst * NH + h]);
    exW[idx] = ex;
    atomicAdd(&segSum[(size_t)dst * NH + h], ex);
}

// ---------------------------------------------------------------------------
// Pass 3: agg[dst] += (v[src]+ve[e]) * alpha.  One wave per edge; 4 f32 atomics
// per lane into the L2-resident agg buffer.
// ---------------------------------------------------------------------------
__global__ void edge_agg_kernel(const float* __restrict__ Vp,
                                const float* __restrict__ EF,
                                const int* __restrict__ esrc,
                                const int* __restrict__ edst,
                                const float* __restrict__ Wve,
                                const float* __restrict__ exW,
                                const float* __restrict__ segSum,
                                float* __restrict__ agg, int E)
{
    __shared__ float WveS[4 * HID];
    for (int i = threadIdx.x; i < 4 * HID; i += blockDim.x) WveS[i] = Wve[i];
    __syncthreads();

    const int wave = threadIdx.x >> 5;
    const int lane = threadIdx.x & 31;
    const int e = blockIdx.x * 8 + wave;
    if (e >= E) return;

    const int h  = lane >> 2;
    const int dq = lane & 3;
    const int c  = h * HD + dq * 4;

    const int src = esrc[e];
    const int dst = edst[e];
    const float alpha = exW[(size_t)e * NH + h]
                      / (segSum[(size_t)dst * NH + h] + 1e-16f);
    const float e0 = EF[e * 4 + 0], e1 = EF[e * 4 + 1];
    const float e2 = EF[e * 4 + 2], e3 = EF[e * 4 + 3];

    const float4 v4 = *(const float4*)(Vp + (size_t)src * HID + c);
#pragma unroll
    for (int j = 0; j < 4; ++j) {
        const float ve = e0 * WveS[c + j] + e1 * WveS[HID + c + j]
                       + e2 * WveS[2 * HID + c + j] + e3 * WveS[3 * HID + c + j];
        atomicAdd(&agg[(size_t)dst * HID + c + j], ((&v4.x)[j] + ve) * alpha);
    }
}

// ---------------------------------------------------------------------------
// Fold BatchNorm into the MLP's first layer:
//   s = gamma*rsqrt(var+eps), t = beta - mean*s
//   W1p[k][c] = s[k]*W1[k][c]  (k < 128);  biasp[c] = b1[c] + sum_k t[k]*W1[k][c]
// ---------------------------------------------------------------------------
__global__ void bn_prep_kernel(const float* __restrict__ colsum,
                               const float* __restrict__ colsq,
                               const float* __restrict__ gamma,
                               const float* __restrict__ beta,
                               const float* __restrict__ W1,
                               const float* __restrict__ b1,
                               float* __restrict__ W1p,
                               float* __restrict__ biasp, int M)
{
    __shared__ float sS[HID], tS[HID];
    const int c = threadIdx.x;
    if (c < HID) {
        const float mean = colsum[c] / (float)M;
        const float var  = colsq[c] / (float)M - mean * mean;
        const float s    = gamma[c] * rsqrtf(var + 1e-5f);
        sS[c] = s;
        tS[c] = beta[c] - mean * s;
    }
    __syncthreads();
    if (c < HID) {
        float acc = b1[c];
        for (int k = 0; k < HID; ++k) {
            const float w = W1[k * HID + c];
            W1p[k * HID + c] = sS[k] * w;
            acc += tS[k] * w;
        }
        biasp[c] = acc;
    }
}

// ---------------------------------------------------------------------------
// Fused MLP:  H = relu(out_bn @ W1a + right @ W1b + biasp)   (BN pre-folded)
//             Y = H @ W2 + b2
// One block per 16-row strip; H tile staged in LDS between the two GEMMs.
// ---------------------------------------------------------------------------
__global__ void mlp_fused_kernel(const float* __restrict__ outp,
                                 const float* __restrict__ right,
                                 const float* __restrict__ W1p,
                                 const float* __restrict__ W1,   // rows 128..255 = W1b
                                 const float* __restrict__ biasp,
                                 const float* __restrict__ W2,
                                 const float* __restrict__ b2,
                                 float* __restrict__ Y, int M)
{
    __shared__ float Hs[16 * HID];
    const int colBase = (threadIdx.x >> 5) * 16;
    const int rowBase = blockIdx.x * 16;
    const int lane = threadIdx.x & 31;
    const int half = lane >> 4;
    const int l15  = lane & 15;

    v8f acc = {};
    acc = wmma_tile_mac(outp,  HID, W1p,            HID, rowBase, colBase, HID, M - 1, acc);
    acc = wmma_tile_mac(right, HID, W1 + HID * HID, HID, rowBase, colBase, HID, M - 1, acc);

    union { v8f v; float f[8]; } u; u.v = acc;
    const float bb = biasp[colBase + l15];
#pragma unroll
    for (int r = 0; r < 8; ++r) {
        float v = u.f[r] + bb;
        v = v > 0.0f ? v : 0.0f;
        Hs[(r + 8 * half) * HID + colBase + l15] = v;
    }
    __syncthreads();

    v8f acc2 = {};
    acc2 = wmma_tile_mac(Hs, HID, W2, HID, 0, colBase, HID, 15, acc2);

    union { v8f v; float f[8]; } u2; u2.v = acc2;
    const float b2v = b2[colBase + l15];
#pragma unroll
    for (int r = 0; r < 8; ++r) {
        const int row = rowBase + r + 8 * half;
        if (row < M) Y[(size_t)row * HID + colBase + l15] = u2.f[r] + b2v;
    }
}

// ---------------------------------------------------------------------------
extern "C" void kernel_launch(void* const* d_in, const int* in_sizes, int n_in,
                              void* d_out, int out_size, void* d_ws, size_t ws_size,
                              hipStream_t stream)
{
    const float* left  = (const float*)d_in[0];
    const float* right = (const float*)d_in[1];
    const float* EF    = (const float*)d_in[2];
    const int*   esrc  = (const int*)d_in[3];
    const int*   edst  = (const int*)d_in[4];
    const float* Wq    = (const float*)d_in[5];
    const float* Wk    = (const float*)d_in[6];
    const float* Wv    = (const float*)d_in[7];
    const float* Wke   = (const float*)d_in[8];
    const float* Wve   = (const float*)d_in[9];
    const float* Wout  = (const float*)d_in[10];
    const float* gamma = (const float*)d_in[11];
    const float* beta  = (const float*)d_in[12];
    const float* W1    = (const float*)d_in[13];
    const float* b1    = (const float*)d_in[14];
    const float* W2    = (const float*)d_in[15];
    const float* b2    = (const float*)d_in[16];
    float* Y = (float*)d_out;

    const int NL = in_sizes[0] / HID;
    const int NR = in_sizes[1] / HID;
    const int E  = in_sizes[3];

    // workspace carve-up (floats)
    float* ws = (float*)d_ws;
    float* Qf      = ws;                   // NR*128
    float* Kf      = Qf + (size_t)NR * HID;        // NL*128
    float* Vf      = Kf + (size_t)NL * HID;        // NL*128
    float* logitsW = Vf + (size_t)NL * HID;        // E*8
    float* exW     = logitsW + (size_t)E * NH;     // E*8
    float* segMax  = exW + (size_t)E * NH;         // NR*8
    float* segSum  = segMax + (size_t)NR * NH;     // NR*8
    float* aggW    = segSum + (size_t)NR * NH;     // NR*128
    float* outW    = aggW + (size_t)NR * HID;      // NR*128
    float* colsum  = outW + (size_t)NR * HID;      // 128
    float* colsq   = colsum + HID;                 // 128
    float* W1p     = colsq + HID;                  // 128*128
    float* biasp   = W1p + HID * HID;              // 128

    const dim3 blk(256);
    const int mTilesR = (NR + 15) / 16;
    const int mTilesL = (NL + 15) / 16;
    const int eBlocks = (E + 7) / 8;

    init_ws_kernel<<<(NR * HID + 255) / 256, blk, 0, stream>>>(
        aggW, segMax, segSum, colsum, colsq, NR);

    // node projections (fp32 WMMA)
    gemm128_wmma<<<mTilesR, blk, 0, stream>>>(right, Wq, Qf, NR, HID, nullptr, nullptr);
    gemm128_wmma<<<mTilesL, blk, 0, stream>>>(left,  Wk, Kf, NL, HID, nullptr, nullptr);
    gemm128_wmma<<<mTilesL, blk, 0, stream>>>(left,  Wv, Vf, NL, HID, nullptr, nullptr);

    // edge attention: logits + segment max
    edge_logits_kernel<<<eBlocks, blk, 0, stream>>>(
        Qf, Kf, EF, esrc, edst, Wke, logitsW, segMax, E);
    // exp + segment sum
    edge_exp_kernel<<<(E * NH + 255) / 256, blk, 0, stream>>>(
        logitsW, edst, segMax, exW, segSum, E);
    // weighted message aggregation
    edge_agg_kernel<<<eBlocks, blk, 0, stream>>>(
        Vf, EF, esrc, edst, Wve, exW, segSum, aggW, E);

    // out-projection + BN batch statistics
    gemm128_wmma<<<mTilesR, blk, 0, stream>>>(aggW, Wout, outW, NR, HID, colsum, colsq);

    // fold BN into MLP layer 1
    bn_prep_kernel<<<1, 128, 0, stream>>>(colsum, colsq, gamma, beta, W1, b1,
                                          W1p, biasp, NR);

    // fused 2-layer MLP (WMMA x3 GEMM chains per tile)
    mlp_fused_kernel<<<mTilesR, blk, 0, stream>>>(
        outW, right, W1p, W1, biasp, W2, b2, Y, NR);
}